// PolyhedronModel_82489141886992
// MI455X (gfx1250) — compile-verified
//
#include <hip/hip_runtime.h>

// ---------------------------------------------------------------------------
// CGConv dual-branch GNN for MI455X (gfx1250, wave32).
// Memory-bound: ~256MB edge streaming dominates (roofline ~11us @ 23.3TB/s).
// Edge linear (z[E,6] @ [Wf|Ws][6,4]) runs on V_WMMA_F32_16X16X4_F32 with
// f32 accumulation (bit-compatible with the f32 reference).
// ---------------------------------------------------------------------------

typedef __attribute__((ext_vector_type(2))) float v2f;
typedef __attribute__((ext_vector_type(8))) float v8f;

__device__ __forceinline__ float sigmoid_f(float x) {
  return 1.0f / (1.0f + expf(-x));
}
__device__ __forceinline__ float softplus_f(float x) {
  // stable log1p(exp(x)) as in jax.nn.softplus
  return fmaxf(x, 0.0f) + log1pf(expf(-fabsf(x)));
}

// ---------------------------------------------------------------------------
__global__ void zero_kernel(float* __restrict__ p, long long n) {
  long long i = (long long)blockIdx.x * blockDim.x + threadIdx.x;
  long long stride = (long long)gridDim.x * blockDim.x;
  for (; i < n; i += stride) p[i] = 0.0f;
}

// ---------------------------------------------------------------------------
// WMMA edge kernel: each wave processes 16 edges per iteration.
//   A (16x4 f32): rows 0..3 = [Wf_c0 | Wf_c1 | Ws_c0 | Ws_c1]^T, rows 4..15 = 0
//     lanes 0-15 hold K=0,1 (VGPR0,1); lanes 16-31 hold K=2,3.
//   B (4x16 f32): column N = edge; lanes 0-15 supply z0,z1 = x[dst],
//     lanes 16-31 supply z2,z3 = x[src]. Second WMMA: z4,z5 = edge_attr (lo),
//     zero padding (hi).
//   D (16x16 f32): lane l (<16) holds edge l's outputs in acc[0..3].
// ---------------------------------------------------------------------------
__global__ void edge_wmma_kernel(const float2* __restrict__ x,
                                 const int* __restrict__ src,
                                 const int* __restrict__ dst,
                                 const float2* __restrict__ ea,
                                 const float* __restrict__ Wf,
                                 const float* __restrict__ bf,
                                 const float* __restrict__ Ws,
                                 const float* __restrict__ bs,
                                 float* __restrict__ agg,
                                 long long nchunk) {
  const int lane = threadIdx.x & 31;
  const int m = lane & 15;
  const bool hi = (lane >= 16);

  // Weight row for output channel m over the 6 z-dims (0 for m >= 4).
  float w[6];
#pragma unroll
  for (int k = 0; k < 6; ++k) {
    float v = 0.0f;
    if (m < 2)      v = Wf[k * 2 + m];        // Wf is (6,2) row-major
    else if (m < 4) v = Ws[k * 2 + (m - 2)];  // Ws is (6,2) row-major
    w[k] = v;
  }
  v2f a1, a2;
  a1.x = hi ? w[2] : w[0];   // VGPR0: K=0 (lo) / K=2 (hi)
  a1.y = hi ? w[3] : w[1];   // VGPR1: K=1 (lo) / K=3 (hi)
  a2.x = hi ? 0.0f : w[4];   // 2nd WMMA: K=0 -> z4 ; hi half = zero pad
  a2.y = hi ? 0.0f : w[5];   //           K=1 -> z5

  const float bf0 = bf[0], bf1 = bf[1];
  const float bs0 = bs[0], bs1 = bs[1];

  long long wid = (((long long)blockIdx.x * blockDim.x) + threadIdx.x) >> 5;
  long long nwaves = ((long long)gridDim.x * blockDim.x) >> 5;

  for (long long c = wid; c < nchunk; c += nwaves) {
    long long e = c * 16 + m;
    int node = hi ? src[e] : dst[e];   // z0,z1 = x_i = x[dst]; z2,z3 = x_j = x[src]
    float2 xv = x[node];
    v2f b1v; b1v.x = xv.x; b1v.y = xv.y;
    v2f b2v;
    if (!hi) {
      float2 ev = ea[e];
      b2v.x = ev.x; b2v.y = ev.y;      // z4,z5 = edge_attr
    } else {
      b2v.x = 0.0f; b2v.y = 0.0f;      // K=6,7 padding
    }
    v8f acc = {0.f, 0.f, 0.f, 0.f, 0.f, 0.f, 0.f, 0.f};
    acc = __builtin_amdgcn_wmma_f32_16x16x4_f32(false, a1, false, b1v,
                                                (short)0, acc, false, false);
    acc = __builtin_amdgcn_wmma_f32_16x16x4_f32(false, a2, false, b2v,
                                                (short)0, acc, false, false);
    if (!hi) {
      float f0 = sigmoid_f(acc[0] + bf0);
      float f1 = sigmoid_f(acc[1] + bf1);
      float s0 = softplus_f(acc[2] + bs0);
      float s1 = softplus_f(acc[3] + bs1);
      atomicAdd(&agg[(long long)node * 2 + 0], f0 * s0);
      atomicAdd(&agg[(long long)node * 2 + 1], f1 * s1);
    }
  }
}

// Scalar fallback for a tail that is not a multiple of 16 edges.
__global__ void edge_scalar_kernel(const float2* __restrict__ x,
                                   const int* __restrict__ src,
                                   const int* __restrict__ dst,
                                   const float2* __restrict__ ea,
                                   const float* __restrict__ Wf,
                                   const float* __restrict__ bf,
                                   const float* __restrict__ Ws,
                                   const float* __restrict__ bs,
                                   float* __restrict__ agg,
                                   long long start, long long E) {
  long long i = start + (long long)blockIdx.x * blockDim.x + threadIdx.x;
  long long stride = (long long)gridDim.x * blockDim.x;
  for (; i < E; i += stride) {
    int d = dst[i], s = src[i];
    float2 xi = x[d], xj = x[s], ev = ea[i];
    float z[6] = {xi.x, xi.y, xj.x, xj.y, ev.x, ev.y};
#pragma unroll
    for (int c = 0; c < 2; ++c) {
      float f = bf[c], g = bs[c];
#pragma unroll
      for (int k = 0; k < 6; ++k) {
        f += z[k] * Wf[k * 2 + c];
        g += z[k] * Ws[k * 2 + c];
      }
      atomicAdd(&agg[(long long)d * 2 + c], sigmoid_f(f) * softplus_f(g));
    }
  }
}

// ---------------------------------------------------------------------------
__global__ void node_kernel(const float2* __restrict__ x,
                            const float* __restrict__ agg,
                            const int* __restrict__ batch,
                            const float* __restrict__ W1,  // (2,3) row-major
                            const float* __restrict__ b1,
                            float* __restrict__ sums,      // (B,3)
                            float* __restrict__ cnt,       // (B)
                            long long N) {
  long long i = (long long)blockIdx.x * blockDim.x + threadIdx.x;
  long long stride = (long long)gridDim.x * blockDim.x;
  for (; i < N; i += stride) {
    float2 xv = x[i];
    float h0 = xv.x + agg[i * 2 + 0];
    float h1 = xv.y + agg[i * 2 + 1];
    int b = batch[i];
#pragma unroll
    for (int k = 0; k < 3; ++k) {
      float t = fmaxf(h0 * W1[k] + h1 * W1[3 + k] + b1[k], 0.0f);
      atomicAdd(&sums[(long long)b * 3 + k], t);
    }
    atomicAdd(&cnt[b], 1.0f);
  }
}

// ---------------------------------------------------------------------------
__global__ void head_kernel(const float* __restrict__ sums_a,
                            const float* __restrict__ cnt_a,
                            const float* __restrict__ sums_b,
                            const float* __restrict__ cnt_b,
                            const float* __restrict__ W2,  // (6,1)
                            const float* __restrict__ b2,
                            const float* __restrict__ targets,
                            float* __restrict__ out,       // (B)
                            float* __restrict__ partials,  // per-block sq-err
                            int B) {
  __shared__ float red[256];
  int b = blockIdx.x * 256 + threadIdx.x;
  float sq = 0.0f;
  if (b < B) {
    float ca = fmaxf(cnt_a[b], 1.0f);
    float cb = fmaxf(cnt_b[b], 1.0f);
    float o = b2[0];
#pragma unroll
    for (int k = 0; k < 3; ++k) {
      o += (sums_a[b * 3 + k] / ca) * W2[k];
      o += (sums_b[b * 3 + k] / cb) * W2[3 + k];
    }
    o = fmaxf(o, 0.0f);
    out[b] = o;
    float d = o - targets[b];
    sq = d * d;
  }
  red[threadIdx.x] = sq;
  __syncthreads();
  for (int s = 128; s > 0; s >>= 1) {
    if ((int)threadIdx.x < s) red[threadIdx.x] += red[threadIdx.x + s];
    __syncthreads();
  }
  if (threadIdx.x == 0) partials[blockIdx.x] = red[0];
}

__global__ void loss_kernel(const float* __restrict__ partials, int nb,
                            float* __restrict__ loss_out, int B) {
  float s = 0.0f;
  for (int i = 0; i < nb; ++i) s += partials[i];
  loss_out[0] = s / (float)B;
}

// ---------------------------------------------------------------------------
extern "C" void kernel_launch(void* const* d_in, const int* in_sizes, int n_in,
                              void* d_out, int out_size, void* d_ws, size_t ws_size,
                              hipStream_t stream) {
  const float2* x_a   = (const float2*)d_in[0];
  const int*    ei_a  = (const int*)d_in[1];
  const float2* ea_a  = (const float2*)d_in[2];
  const int*    bat_a = (const int*)d_in[3];
  const float2* x_b   = (const float2*)d_in[4];
  const int*    ei_b  = (const int*)d_in[5];
  const float2* ea_b  = (const float2*)d_in[6];
  const int*    bat_b = (const int*)d_in[7];
  const float*  tgt   = (const float*)d_in[8];
  const float*  Wf    = (const float*)d_in[10];
  const float*  bf    = (const float*)d_in[11];
  const float*  Ws    = (const float*)d_in[12];
  const float*  bs    = (const float*)d_in[13];
  const float*  W1    = (const float*)d_in[14];
  const float*  b1    = (const float*)d_in[15];
  const float*  W2    = (const float*)d_in[16];
  const float*  b2    = (const float*)d_in[17];

  const long long N = in_sizes[3];       // nodes
  const long long E = in_sizes[1] / 2;   // edges (edge_index is (2,E))
  const int       B = in_sizes[8];       // graphs

  const int* src_a = ei_a;
  const int* dst_a = ei_a + E;
  const int* src_b = ei_b;
  const int* dst_b = ei_b + E;

  // workspace layout (floats)
  float* ws     = (float*)d_ws;
  float* agg_a  = ws;                    // 2N
  float* agg_b  = agg_a + 2 * N;         // 2N
  float* sums_a = agg_b + 2 * N;         // 3B
  float* cnt_a  = sums_a + 3 * (long long)B;  // B
  float* sums_b = cnt_a + B;             // 3B
  float* cnt_b  = sums_b + 3 * (long long)B;  // B
  float* parts  = cnt_b + B;             // up to 4096

  // 1) zero accumulators
  long long nz = 4 * N + 8 * (long long)B;
  int zb = (int)((nz + 255) / 256);
  zero_kernel<<<zb, 256, 0, stream>>>(ws, nz);

  // 2) edge aggregation (WMMA main body + scalar tail)
  long long nchunk = E / 16;
  long long emain  = nchunk * 16;
  if (nchunk > 0) {
    edge_wmma_kernel<<<2048, 256, 0, stream>>>(x_a, src_a, dst_a, ea_a,
                                               Wf, bf, Ws, bs, agg_a, nchunk);
    edge_wmma_kernel<<<2048, 256, 0, stream>>>(x_b, src_b, dst_b, ea_b,
                                               Wf, bf, Ws, bs, agg_b, nchunk);
  }
  if (emain < E) {
    edge_scalar_kernel<<<64, 256, 0, stream>>>(x_a, src_a, dst_a, ea_a,
                                               Wf, bf, Ws, bs, agg_a, emain, E);
    edge_scalar_kernel<<<64, 256, 0, stream>>>(x_b, src_b, dst_b, ea_b,
                                               Wf, bf, Ws, bs, agg_b, emain, E);
  }

  // 3) node update + mean-pool accumulation
  int nb = (int)((N + 255) / 256);
  node_kernel<<<nb, 256, 0, stream>>>(x_a, agg_a, bat_a, W1, b1, sums_a, cnt_a, N);
  node_kernel<<<nb, 256, 0, stream>>>(x_b, agg_b, bat_b, W1, b1, sums_b, cnt_b, N);

  // 4) head + deterministic loss reduction
  int hb = (B + 255) / 256;
  head_kernel<<<hb, 256, 0, stream>>>(sums_a, cnt_a, sums_b, cnt_b, W2, b2, tgt,
                                      (float*)d_out, parts, B);
  loss_kernel<<<1, 1, 0, stream>>>(parts, hb, (float*)d_out + B, B);
}